// TasteGNN_50689204027704
// MI455X (gfx1250) — compile-verified
//
#include <hip/hip_runtime.h>
#include <math.h>

typedef __attribute__((ext_vector_type(2))) float v2f;
typedef __attribute__((ext_vector_type(8))) float v8f;

#define D 128
#define NI 100000
#define NT 50000
#define NE 500000

// ---------------------------------------------------------------------------
// GEMM: H[M,128] = X[M,128] @ W[128,128] + bias, fp32 via V_WMMA_F32_16X16X4_F32
// Block: 256 threads = 8 waves. Block covers 128 rows; wave w owns rows
// [row0+16w, row0+16w+16) and accumulates all 8 column tiles (8 x v8f accs).
// W staged in LDS interleaved by k-pairs: lw[(k>>1)*256 + c*2 + (k&1)] so a
// B fragment (b.x=W[k][c], b.y=W[k+1][c]) is one 8-byte LDS load per lane.
// ---------------------------------------------------------------------------
__global__ __launch_bounds__(256) void gemm128_wmma(
    const float* __restrict__ X, const float* __restrict__ W,
    const float* __restrict__ bias, float* __restrict__ H, int M)
{
  __shared__ float lw[64 * 256];  // 65536 B

  for (int i = threadIdx.x; i < D * D; i += 256) {
    int k = i >> 7;
    int c = i & 127;
    lw[(k >> 1) * 256 + (c << 1) + (k & 1)] = W[i];
  }
  __syncthreads();

  const int lane = threadIdx.x & 31;
  const int wv   = threadIdx.x >> 5;
  const int half = lane >> 4;    // 0: K0/K1 (and rows M=r), 1: K2/K3 (rows M=8+r)
  const int mn   = lane & 15;    // A row / B col within 16-tile
  const int rowb = blockIdx.x * 128 + wv * 16;
  if (rowb >= M) return;         // M is a multiple of 16, per-wave all-or-nothing

  v8f acc[8] = {};

  const float* Ap = X + (size_t)(rowb + mn) * D + (half << 1);
  #pragma unroll 4
  for (int k0 = 0; k0 < D; k0 += 4) {
    v2f a = *(const v2f*)(Ap + k0);   // A[m][k0+2h], A[m][k0+2h+1]
    const float* Bp = &lw[(k0 >> 1) * 256 + (half << 8) + (mn << 1)];
    #pragma unroll
    for (int t = 0; t < 8; ++t) {
      v2f b = *(const v2f*)(Bp + (t << 5));  // W[k0+2h][16t+n], W[k0+2h+1][16t+n]
      acc[t] = __builtin_amdgcn_wmma_f32_16x16x4_f32(
          false, a, false, b, (short)0, acc[t], false, false);
    }
  }

  #pragma unroll
  for (int t = 0; t < 8; ++t) {
    const int col = (t << 4) + mn;
    const float bv = bias[col];
    #pragma unroll
    for (int r = 0; r < 8; ++r) {
      // C/D layout: VGPR r -> M=r (lanes 0-15), M=8+r (lanes 16-31), N=lane%16
      H[(size_t)(rowb + r + (half << 3)) * D + col] = acc[t][r] + bv;
    }
  }
}

// a[row] = dot(H[row,:], att) ; one wave (32 lanes x float4) per row
__global__ __launch_bounds__(256) void rowdot(
    const float* __restrict__ H, const float* __restrict__ att,
    float* __restrict__ out, int M)
{
  int row  = blockIdx.x * 8 + (threadIdx.x >> 5);
  int lane = threadIdx.x & 31;
  if (row >= M) return;
  float4 h = ((const float4*)(H + (size_t)row * D))[lane];
  float4 a = ((const float4*)att)[lane];
  float s = h.x * a.x + h.y * a.y + h.z * a.z + h.w * a.w;
  #pragma unroll
  for (int o = 16; o > 0; o >>= 1) s += __shfl_down(s, o, 32);
  if (lane == 0) out[row] = s;
}

// alpha = leaky_relu(a_src[src]+a_dst[dst], 0.2); ex = exp(alpha); z[dst] += ex
// (max-subtraction dropped: softmax is shift-invariant and alpha is O(1))
__global__ __launch_bounds__(256) void edge_alpha(
    const int* __restrict__ esrc, const int* __restrict__ edst,
    const float* __restrict__ as, const float* __restrict__ ad,
    float* __restrict__ ex, float* __restrict__ z, int E)
{
  int e = blockIdx.x * 256 + threadIdx.x;
  if (e >= E) return;
  int d = edst[e];
  float al = as[esrc[e]] + ad[d];
  al = (al > 0.0f) ? al : 0.2f * al;
  float v = expf(al);
  ex[e] = v;
  atomicAdd(&z[d], v);
}

// agg[dst,:] += h_i[src,:] * (ex[e] / (z[dst]+1e-16)) ; 32 lanes x float4 per edge
__global__ __launch_bounds__(256) void edge_scatter(
    const int* __restrict__ esrc, const int* __restrict__ edst,
    const float* __restrict__ ex, const float* __restrict__ z,
    const float* __restrict__ Hi, float* __restrict__ agg, int E)
{
  int e    = blockIdx.x * 8 + (threadIdx.x >> 5);
  int lane = threadIdx.x & 31;
  if (e >= E) return;
  int s = esrc[e], t = edst[e];
  float w = ex[e] / (z[t] + 1e-16f);
  float4 h = ((const float4*)(Hi + (size_t)s * D))[lane];
  float* ap = agg + (size_t)t * D + lane * 4;
  atomicAdd(ap + 0, h.x * w);
  atomicAdd(ap + 1, h.y * w);
  atomicAdd(ap + 2, h.z * w);
  atomicAdd(ap + 3, h.w * w);
}

// T = relu(agg) + x_taste; accumulate per-column sum / sumsq for BatchNorm
__global__ __launch_bounds__(256) void resid_stats(
    const float* __restrict__ agg, const float* __restrict__ xt,
    float* __restrict__ T, float* __restrict__ colsum,
    float* __restrict__ colsq, int Nt)
{
  int c    = threadIdx.x & 127;
  int rsub = threadIdx.x >> 7;   // 0 or 1
  int row0 = blockIdx.x * 32;
  float s = 0.0f, q = 0.0f;
  for (int i = rsub; i < 32; i += 2) {
    int r = row0 + i;
    if (r < Nt) {
      size_t idx = (size_t)r * D + c;
      float a = agg[idx];
      a = a > 0.0f ? a : 0.0f;
      float v = a + xt[idx];
      T[idx] = v;
      s += v;
      q += v * v;
    }
  }
  __shared__ float ls[256], lq[256];
  ls[threadIdx.x] = s;
  lq[threadIdx.x] = q;
  __syncthreads();
  if (rsub == 0) {
    s += ls[threadIdx.x + 128];
    q += lq[threadIdx.x + 128];
    atomicAdd(&colsum[c], s);
    atomicAdd(&colsq[c], q);
  }
}

// fold BN into per-column scale/shift: scale=gamma*rsqrt(var+eps), shift=beta-mean*scale
__global__ void bn_coef(
    const float* __restrict__ colsum, const float* __restrict__ colsq,
    const float* __restrict__ gamma, const float* __restrict__ beta,
    float* __restrict__ scale, float* __restrict__ shift, float invN)
{
  int c = threadIdx.x;
  float mean = colsum[c] * invN;
  float var  = colsq[c] * invN - mean * mean;  // biased var, as in reference
  float sc   = gamma[c] * rsqrtf(var + 1e-5f);
  scale[c] = sc;
  shift[c] = beta[c] - mean * sc;
}

__global__ __launch_bounds__(256) void bn_apply(
    float* __restrict__ T, const float* __restrict__ scale,
    const float* __restrict__ shift, long n)
{
  long i = (long)blockIdx.x * 256 + threadIdx.x;
  if (i >= n) return;
  int c = (int)(i & (D - 1));
  float v = T[i] * scale[c] + shift[c];
  T[i] = v > 0.0f ? v : 0.0f;
}

extern "C" void kernel_launch(void* const* d_in, const int* in_sizes, int n_in,
                              void* d_out, int out_size, void* d_ws, size_t ws_size,
                              hipStream_t stream)
{
  const float* x_ing   = (const float*)d_in[0];
  const float* x_taste = (const float*)d_in[1];
  const float* W_ing   = (const float*)d_in[2];
  const float* b_ing   = (const float*)d_in[3];
  const float* W_taste = (const float*)d_in[4];
  const float* b_taste = (const float*)d_in[5];
  const float* att_src = (const float*)d_in[6];
  const float* att_dst = (const float*)d_in[7];
  const float* gamma   = (const float*)d_in[8];
  const float* beta    = (const float*)d_in[9];
  const int*   esrc    = (const int*)d_in[10];
  const int*   edst    = (const int*)d_in[11];

  // workspace carve-out (256B-aligned)
  char* ws = (char*)d_ws;
  size_t off = 0;
  auto carve = [&](size_t bytes) -> float* {
    float* p = (float*)(ws + off);
    off = (off + bytes + 255) & ~(size_t)255;
    return p;
  };
  float* h_i   = carve((size_t)NI * D * 4);
  float* h_t   = carve((size_t)NT * D * 4);
  float* a_src = carve((size_t)NI * 4);
  float* a_dst = carve((size_t)NT * 4);
  float* ex    = carve((size_t)NE * 4);
  float* z     = carve((size_t)NT * 4);
  float* agg   = carve((size_t)NT * D * 4);
  float* stats = carve((size_t)4 * D * 4);
  float* colsum = stats;
  float* colsq  = stats + D;
  float* scale  = stats + 2 * D;
  float* shift  = stats + 3 * D;

  float* outI = (float*)d_out;                 // [NI, D] passthrough
  float* outT = outI + (size_t)NI * D;         // [NT, D]

  // zero accumulators every call (deterministic, graph-capturable)
  hipMemsetAsync(z, 0, (size_t)NT * 4, stream);
  hipMemsetAsync(agg, 0, (size_t)NT * D * 4, stream);
  hipMemsetAsync(stats, 0, (size_t)2 * D * 4, stream);
  hipMemcpyAsync(outI, x_ing, (size_t)NI * D * 4, hipMemcpyDeviceToDevice, stream);

  gemm128_wmma<<<(NI + 127) / 128, 256, 0, stream>>>(x_ing, W_ing, b_ing, h_i, NI);
  gemm128_wmma<<<(NT + 127) / 128, 256, 0, stream>>>(x_taste, W_taste, b_taste, h_t, NT);
  rowdot<<<(NI + 7) / 8, 256, 0, stream>>>(h_i, att_src, a_src, NI);
  rowdot<<<(NT + 7) / 8, 256, 0, stream>>>(h_t, att_dst, a_dst, NT);
  edge_alpha<<<(NE + 255) / 256, 256, 0, stream>>>(esrc, edst, a_src, a_dst, ex, z, NE);
  edge_scatter<<<(NE + 7) / 8, 256, 0, stream>>>(esrc, edst, ex, z, h_i, agg, NE);
  resid_stats<<<(NT + 31) / 32, 256, 0, stream>>>(agg, x_taste, outT, colsum, colsq, NT);
  bn_coef<<<1, 128, 0, stream>>>(colsum, colsq, gamma, beta, scale, shift, 1.0f / NT);
  bn_apply<<<((long)NT * D + 255) / 256, 256, 0, stream>>>(outT, scale, shift, (long)NT * D);
}